// GraphNeuralNetwork_75806172774423
// MI455X (gfx1250) — compile-verified
//
#include <hip/hip_runtime.h>
#include <math.h>

#define N_NODES 50000
#define N_EDGES 800000
#define EP (N_EDGES + N_NODES)   // edges incl. self loops = 850000
#define FDIM 128
#define NGRAPH 64
#define HID 32
#define LN_EPS 1e-5f
#define XT_PITCH (FDIM + 4)      // LDS row pitch (floats): 528B -> conflict-free A reads

typedef float v2f __attribute__((ext_vector_type(2)));
typedef float v8f __attribute__((ext_vector_type(8)));

// int4 vector + address-space-qualified pointee types for the async builtin
typedef int v4i __attribute__((vector_size(16)));
typedef __attribute__((address_space(1))) v4i as1_v4i;   // global
typedef __attribute__((address_space(3))) v4i as3_v4i;   // LDS

#if defined(__has_builtin)
#if __has_builtin(__builtin_amdgcn_global_load_async_to_lds_b128)
#define HAVE_ASYNC_LDS 1
#endif
#endif

__device__ __forceinline__ void wait_asynccnt0() {
#if defined(__has_builtin) && __has_builtin(__builtin_amdgcn_s_wait_asynccnt)
    __builtin_amdgcn_s_wait_asynccnt(0);
#else
    asm volatile("s_wait_asynccnt 0" ::: "memory");
#endif
}

// ---- order-preserving float<->uint mapping for atomic max on floats ----
__device__ __forceinline__ unsigned ford(float f) {
    unsigned u = __float_as_uint(f);
    return (u & 0x80000000u) ? ~u : (u | 0x80000000u);
}
__device__ __forceinline__ float funord(unsigned u) {
    return (u & 0x80000000u) ? __uint_as_float(u & 0x7fffffffu) : __uint_as_float(~u);
}
#define NEG_INF_ORD 0x007FFFFFu   // ford(-inf)

__device__ __forceinline__ void edge_sd(int e, const int* ei, int& src, int& dst) {
    if (e < N_EDGES) { src = ei[e]; dst = ei[N_EDGES + e]; }
    else             { src = dst = e - N_EDGES; }          // self loop
}

// -------------------- misc fills --------------------
__global__ void k_zero_f(float* p, int n) {
    int t = blockIdx.x * blockDim.x + threadIdx.x;
    if (t < n) p[t] = 0.f;
}
__global__ void k_fill_u(unsigned* p, unsigned v, int n) {
    int t = blockIdx.x * blockDim.x + threadIdx.x;
    if (t < n) p[t] = v;
}

// -------------------- self-loop attr (fill_value='mean') --------------------
__global__ void k_count_sum(const int* ei, const float* eattr, float* cnt, float* lsum) {
    int e = blockIdx.x * blockDim.x + threadIdx.x;
    if (e >= N_EDGES) return;
    int dst = ei[N_EDGES + e];
    atomicAdd(&cnt[dst], 1.f);
    atomicAdd(&lsum[dst * 3 + 0], eattr[e * 3 + 0]);
    atomicAdd(&lsum[dst * 3 + 1], eattr[e * 3 + 1]);
    atomicAdd(&lsum[dst * 3 + 2], eattr[e * 3 + 2]);
}
__global__ void k_loop_attr(const float* cnt, float* lsum) {
    int n = blockIdx.x * blockDim.x + threadIdx.x;
    if (n >= N_NODES) return;
    float d = 1.f / fmaxf(cnt[n], 1.f);
    lsum[n * 3 + 0] *= d;
    lsum[n * 3 + 1] *= d;
    lsum[n * 3 + 2] *= d;
}

// -------------------- per-layer init: acc=bo, smax=-inf, den=0 --------------------
__global__ void k_init_layer(float* acc, const float* bo, unsigned* smax, float* den) {
    int t = blockIdx.x * blockDim.x + threadIdx.x;
    if (t < N_NODES * FDIM) acc[t] = bo[t & (FDIM - 1)];
    if (t < N_NODES * 4) { smax[t] = NEG_INF_ORD; den[t] = 0.f; }
}

// -------------------- fp32 WMMA GEMM: xl = X@Wl^T + bl, xr = X@Wr^T + br ----------
// block = 512 threads = 16 waves. waves 0..7 -> Wl col tiles 0..7; 8..15 -> Wr.
// The 16x128 X tile is staged once per block into LDS via the CDNA5 async
// global->LDS data mover (ASYNCcnt), then all 16 waves read their A fragments
// from LDS (conflict-free via padded pitch) while streaming B from L2.
__global__ void k_gemm_wmma(const float* __restrict__ X,
                            const float* __restrict__ Wl, const float* __restrict__ bl,
                            const float* __restrict__ Wr, const float* __restrict__ br,
                            float* __restrict__ xl, float* __restrict__ xr) {
    __shared__ float xt[16 * XT_PITCH];     // 16 rows x 132 floats = 8.25 KB

    int rowbase = blockIdx.x * 16;

    // ---- async stage: thread t copies 16B of row (t>>5), cols 4*(t&31) ----
    {
        int t = threadIdx.x;                // 0..511
        int trow = t >> 5;
        int c4 = (t & 31) * 4;
        int grow = rowbase + trow;
        if (grow >= N_NODES) grow = N_NODES - 1;   // clamp (stores masked later)
        const float* gsrc = X + (size_t)grow * FDIM + c4;
        float* ldst = &xt[trow * XT_PITCH + c4];
#ifdef HAVE_ASYNC_LDS
        __builtin_amdgcn_global_load_async_to_lds_b128(
            (as1_v4i*)(void*)gsrc, (as3_v4i*)(void*)ldst, 0, 0);
        wait_asynccnt0();
#else
        *(float4*)ldst = *(const float4*)gsrc;
#endif
    }
    __syncthreads();

    int wave = threadIdx.x >> 5;
    int lane = threadIdx.x & 31;
    int sel = wave >> 3;          // 0 -> left, 1 -> right
    int coltile = wave & 7;       // 8 tiles of 16 features
    const float* W = sel ? Wr : Wl;
    const float* bias = sel ? br : bl;
    float* out = sel ? xr : xl;

    int lo = lane & 15;
    int hi = lane >> 4;
    int wrow = coltile * 16 + lo;               // global output feature for B frag

    const float* arow = &xt[lo * XT_PITCH];     // A row in LDS
    const float* wrowp = W + (size_t)wrow * FDIM;

    v8f acc = {0.f, 0.f, 0.f, 0.f, 0.f, 0.f, 0.f, 0.f};
#pragma unroll
    for (int kc = 0; kc < FDIM; kc += 4) {
        int k = kc + 2 * hi;
        v2f a = *(const v2f*)(arow + k);    // A[m=lane&15][k], A[m][k+1]  (ds_load_b64)
        v2f b = *(const v2f*)(wrowp + k);   // B[k][n=lane&15] = W[n][k]
        acc = __builtin_amdgcn_wmma_f32_16x16x4_f32(
            /*neg_a=*/false, a, /*neg_b=*/false, b,
            /*c_mod=*/(short)0, acc, /*reuse_a=*/false, /*reuse_b=*/false);
    }

    int col = coltile * 16 + lo;
    float bv = bias[col];
#pragma unroll
    for (int v = 0; v < 8; v++) {
        int r = rowbase + v + 8 * hi;       // C layout: vgpr v, lane>=16 -> M=v+8
        if (r < N_NODES) out[(size_t)r * FDIM + col] = acc[v] + bv;
    }
}

// -------------------- edge pass 1: attention logits + segment max -----------------
// one wave per edge; lane handles 4 channels; 8-lane groups = one head
__global__ void k_edge_score(const int* __restrict__ ei, const float* __restrict__ eattr,
                             const float* __restrict__ lattr,
                             const float* __restrict__ xl, const float* __restrict__ xr,
                             const float* __restrict__ We, const float* __restrict__ att,
                             float* __restrict__ sbuf, unsigned* __restrict__ smax) {
    int wid = (blockIdx.x * blockDim.x + threadIdx.x) >> 5;
    int lane = threadIdx.x & 31;
    if (wid >= EP) return;
    int src, dst; edge_sd(wid, ei, src, dst);
    const float* ap = (wid < N_EDGES) ? (eattr + (size_t)wid * 3)
                                      : (lattr + (size_t)(wid - N_EDGES) * 3);
    float ea0 = ap[0], ea1 = ap[1], ea2 = ap[2];
    int c4 = lane * 4;
    float4 vl = *(const float4*)(xl + (size_t)src * FDIM + c4);
    float4 vr = *(const float4*)(xr + (size_t)dst * FDIM + c4);
    float mv[4] = {vl.x + vr.x, vl.y + vr.y, vl.z + vr.z, vl.w + vr.w};
    float s = 0.f;
#pragma unroll
    for (int j = 0; j < 4; j++) {
        int c = c4 + j;
        float ev = ea0 * We[c * 3 + 0] + ea1 * We[c * 3 + 1] + ea2 * We[c * 3 + 2];
        float m = mv[j] + ev;
        m = (m > 0.f) ? m : 0.2f * m;       // leaky_relu(0.2)
        s += m * att[c];                    // att (H,C) flat == global channel index
    }
    s += __shfl_xor(s, 4, 8);
    s += __shfl_xor(s, 2, 8);
    s += __shfl_xor(s, 1, 8);
    if ((lane & 7) == 0) {
        int h = lane >> 3;
        sbuf[(size_t)wid * 4 + h] = s;
        atomicMax(&smax[dst * 4 + h], ford(s));
    }
}

// -------------------- edge pass 2: exp(s - smax), segment sum ---------------------
__global__ void k_edge_norm(const int* __restrict__ ei, float* __restrict__ sbuf,
                            const unsigned* __restrict__ smax, float* __restrict__ den) {
    int t = blockIdx.x * blockDim.x + threadIdx.x;
    if (t >= EP * 4) return;
    int e = t >> 2, h = t & 3;
    int src, dst; edge_sd(e, ei, src, dst);
    (void)src;
    float ex = __expf(sbuf[t] - funord(smax[dst * 4 + h]));
    sbuf[t] = ex;
    atomicAdd(&den[dst * 4 + h], ex);
}

// -------------------- edge pass 3: acc[dst] += alpha * xl[src] --------------------
__global__ void k_edge_aggr(const int* __restrict__ ei, const float* __restrict__ sbuf,
                            const float* __restrict__ den, const float* __restrict__ xl,
                            float* __restrict__ acc) {
    int wid = (blockIdx.x * blockDim.x + threadIdx.x) >> 5;
    int lane = threadIdx.x & 31;
    if (wid >= EP) return;
    int src, dst; edge_sd(wid, ei, src, dst);
    int h = lane >> 3;
    float alpha = sbuf[(size_t)wid * 4 + h] / (den[dst * 4 + h] + 1e-16f);
    int c4 = lane * 4;
    float4 vl = *(const float4*)(xl + (size_t)src * FDIM + c4);
    float* dp = acc + (size_t)dst * FDIM + c4;
    atomicAdd(dp + 0, alpha * vl.x);
    atomicAdd(dp + 1, alpha * vl.y);
    atomicAdd(dp + 2, alpha * vl.z);
    atomicAdd(dp + 3, alpha * vl.w);
}

// -------------------- relu (+ layernorm) : one wave per node ----------------------
__global__ void k_post(const float* __restrict__ acc, const float* __restrict__ g,
                       const float* __restrict__ b, float* __restrict__ out, int do_ln) {
    int wid = (blockIdx.x * blockDim.x + threadIdx.x) >> 5;
    int lane = threadIdx.x & 31;
    if (wid >= N_NODES) return;
    int c4 = lane * 4;
    float4 v = *(const float4*)(acc + (size_t)wid * FDIM + c4);
    float r0 = fmaxf(v.x, 0.f), r1 = fmaxf(v.y, 0.f);
    float r2 = fmaxf(v.z, 0.f), r3 = fmaxf(v.w, 0.f);
    if (do_ln) {
        float ps = r0 + r1 + r2 + r3;
#pragma unroll
        for (int off = 16; off >= 1; off >>= 1) ps += __shfl_xor(ps, off, 32);
        float mu = ps * (1.f / FDIM);
        float d0 = r0 - mu, d1 = r1 - mu, d2 = r2 - mu, d3 = r3 - mu;
        float pv = d0 * d0 + d1 * d1 + d2 * d2 + d3 * d3;
#pragma unroll
        for (int off = 16; off >= 1; off >>= 1) pv += __shfl_xor(pv, off, 32);
        float inv = rsqrtf(pv * (1.f / FDIM) + LN_EPS);
        r0 = d0 * inv * g[c4 + 0] + b[c4 + 0];
        r1 = d1 * inv * g[c4 + 1] + b[c4 + 1];
        r2 = d2 * inv * g[c4 + 2] + b[c4 + 2];
        r3 = d3 * inv * g[c4 + 3] + b[c4 + 3];
    }
    *(float4*)(out + (size_t)wid * FDIM + c4) = make_float4(r0, r1, r2, r3);
}

// -------------------- graph max pool --------------------
__global__ void k_pool(const float* __restrict__ xbuf, const int* __restrict__ batch,
                       unsigned* __restrict__ pu) {
    int t = blockIdx.x * blockDim.x + threadIdx.x;
    if (t >= N_NODES * FDIM) return;
    int n = t >> 7, c = t & (FDIM - 1);
    atomicMax(&pu[batch[n] * FDIM + c], ford(xbuf[t]));
}

// -------------------- tiny MLP head: 128 -> 32 -> 32 -> 32 -> 1 -------------------
__global__ void k_mlp(const unsigned* __restrict__ pu,
                      const float* pw1, const float* pb1,
                      const float* pw2, const float* pb2,
                      const float* pw3, const float* pb3,
                      const float* pw4, const float* pb4,
                      float* __restrict__ out) {
    int gidx = threadIdx.x;
    if (gidx >= NGRAPH) return;
    float xv[FDIM];
    for (int c = 0; c < FDIM; c++) {
        float f = funord(pu[gidx * FDIM + c]);
        xv[c] = isfinite(f) ? f : 0.f;      // empty-graph guard
    }
    float h1[HID];
    for (int o = 0; o < HID; o++) {
        float a = pb1[o];
        for (int k = 0; k < FDIM; k++) a += pw1[o * FDIM + k] * xv[k];
        h1[o] = fmaxf(a, 0.f);
    }
    float h2[HID];
    for (int o = 0; o < HID; o++) {
        float a = pb2[o];
        for (int k = 0; k < HID; k++) a += pw2[o * HID + k] * h1[k];
        h2[o] = fmaxf(a, 0.f);
    }
    float h3[HID];
    for (int o = 0; o < HID; o++) {
        float a = pb3[o];
        for (int k = 0; k < HID; k++) a += pw3[o * HID + k] * h2[k];
        h3[o] = fmaxf(a, 0.f);
    }
    float o = pb4[0];
    for (int k = 0; k < HID; k++) o += pw4[k] * h3[k];
    out[gidx] = o;
}

// ==================================================================================
extern "C" void kernel_launch(void* const* d_in, const int* in_sizes, int n_in,
                              void* d_out, int out_size, void* d_ws, size_t ws_size,
                              hipStream_t stream) {
    const float* x0    = (const float*)d_in[0];
    const int*   ei    = (const int*)d_in[1];
    const int*   batch = (const int*)d_in[2];
    const float* eattr = (const float*)d_in[3];
    const float* lng[2] = {(const float*)d_in[25], (const float*)d_in[27]};
    const float* lnb[2] = {(const float*)d_in[26], (const float*)d_in[28]};

    float* out = (float*)d_out;
    float* emb = out;                       // [N,128] embedding; reused as accumulator
    float* mlp_out = out + (size_t)N_NODES * FDIM;

    float* ws = (float*)d_ws;
    float* xl   = ws;
    float* xr   = xl + (size_t)N_NODES * FDIM;
    float* xbuf = xr + (size_t)N_NODES * FDIM;
    float* sbuf = xbuf + (size_t)N_NODES * FDIM;          // [EP,4]
    unsigned* smax = (unsigned*)(sbuf + (size_t)EP * 4);  // [N,4]
    float* den  = (float*)(smax + (size_t)N_NODES * 4);   // [N,4]
    float* cnt  = den + (size_t)N_NODES * 4;              // [N]
    float* lsum = cnt + N_NODES;                          // [N,3] -> becomes loop_attr
    unsigned* pu = (unsigned*)(lsum + (size_t)N_NODES * 3); // [G,128]

    const int TB = 256;
    // self-loop edge attributes (mean of incoming edge_attr)
    k_zero_f<<<(N_NODES * 4 + TB - 1) / TB, TB, 0, stream>>>(cnt, N_NODES * 4);
    k_count_sum<<<(N_EDGES + TB - 1) / TB, TB, 0, stream>>>(ei, eattr, cnt, lsum);
    k_loop_attr<<<(N_NODES + TB - 1) / TB, TB, 0, stream>>>(cnt, lsum);

    int ewblocks = (EP + 7) / 8;            // one wave32 per edge, 8 waves/block

    for (int l = 0; l < 3; l++) {
        const float* Wl  = (const float*)d_in[4 + 7 * l + 0];
        const float* bl  = (const float*)d_in[4 + 7 * l + 1];
        const float* Wr  = (const float*)d_in[4 + 7 * l + 2];
        const float* br  = (const float*)d_in[4 + 7 * l + 3];
        const float* We  = (const float*)d_in[4 + 7 * l + 4];
        const float* att = (const float*)d_in[4 + 7 * l + 5];
        const float* bo  = (const float*)d_in[4 + 7 * l + 6];
        const float* xin = (l == 0) ? x0 : xbuf;

        k_init_layer<<<(N_NODES * FDIM + TB - 1) / TB, TB, 0, stream>>>(emb, bo, smax, den);
        k_gemm_wmma<<<(N_NODES + 15) / 16, 512, 0, stream>>>(xin, Wl, bl, Wr, br, xl, xr);
        k_edge_score<<<ewblocks, TB, 0, stream>>>(ei, eattr, lsum, xl, xr, We, att, sbuf, smax);
        k_edge_norm<<<(EP * 4 + TB - 1) / TB, TB, 0, stream>>>(ei, sbuf, smax, den);
        k_edge_aggr<<<ewblocks, TB, 0, stream>>>(ei, sbuf, den, xl, emb);
        k_post<<<(N_NODES + 7) / 8, TB, 0, stream>>>(
            emb, (l < 2) ? lng[l] : nullptr, (l < 2) ? lnb[l] : nullptr, xbuf, (l < 2) ? 1 : 0);
    }

    // pooling + head
    k_fill_u<<<(NGRAPH * FDIM + TB - 1) / TB, TB, 0, stream>>>(pu, NEG_INF_ORD, NGRAPH * FDIM);
    k_pool<<<(N_NODES * FDIM + TB - 1) / TB, TB, 0, stream>>>(xbuf, batch, pu);
    k_mlp<<<1, 64, 0, stream>>>(pu,
                                (const float*)d_in[29], (const float*)d_in[30],
                                (const float*)d_in[31], (const float*)d_in[32],
                                (const float*)d_in[33], (const float*)d_in[34],
                                (const float*)d_in[35], (const float*)d_in[36],
                                mlp_out);
}